// RotaryPositionalEmbedding_68332929679843
// MI455X (gfx1250) — compile-verified
//
#include <hip/hip_runtime.h>
#include <stdint.h>

// RoPE: out[..., s, :] = R[pos[s]] @ x[..., s, :], where R is block-diagonal
// 2x2 rotations. Bandwidth-bound: gather cos/sin diagonals of R into LDS via
// the CDNA5 async-to-LDS copy path, then stream x/out with NT cache policy.

typedef float v4f __attribute__((ext_vector_type(4)));

#define BH_TOTAL 64        // BATCH * N_HEADS = 4 * 16
#define SEQ 4096
#define DK 128
#define POS_PER_BLOCK 8
#define THREADS 256        // 8 wave32 waves

#if __has_builtin(__builtin_amdgcn_global_load_async_to_lds_b32)
#define HAVE_ASYNC_LDS 1
#else
#define HAVE_ASYNC_LDS 0
#endif

__global__ __launch_bounds__(THREADS)
void rope_apply_kernel(const float* __restrict__ x,
                       const int* __restrict__ token_positions,
                       const float* __restrict__ R,
                       float* __restrict__ out)
{
    __shared__ float s_cos[POS_PER_BLOCK][64];
    __shared__ float s_sin[POS_PER_BLOCK][64];

    const int t = threadIdx.x;

    // ---- Phase 1: gather cos/sin diagonals of R[pos] into LDS ----------
    // cos_k = R[p, 2k, 2k]     -> element offset 258*k
    // sin_k = R[p, 2k+1, 2k]   -> element offset 258*k + 128
#pragma unroll
    for (int e = 0; e < 2; ++e) {
        const int gi = t + THREADS * e;          // 0..511
        const int pl = gi >> 6;                  // local position 0..7
        const int k  = gi & 63;                  // rotation pair index
        const int s  = blockIdx.x * POS_PER_BLOCK + pl;
        const int p  = token_positions[s];
        const float* base = R + (size_t)p * (DK * DK) + 258 * (size_t)k;
#if HAVE_ASYNC_LDS
        // CDNA5 async global->LDS copy (ASYNCcnt-tracked DMA path).
        __builtin_amdgcn_global_load_async_to_lds_b32(
            (__attribute__((address_space(1))) int*)(uintptr_t)base,
            (__attribute__((address_space(3))) int*)&s_cos[pl][k], 0, 0);
        __builtin_amdgcn_global_load_async_to_lds_b32(
            (__attribute__((address_space(1))) int*)(uintptr_t)(base + 128),
            (__attribute__((address_space(3))) int*)&s_sin[pl][k], 0, 0);
#else
        s_cos[pl][k] = base[0];
        s_sin[pl][k] = base[128];
#endif
    }

#if HAVE_ASYNC_LDS
#if __has_builtin(__builtin_amdgcn_s_wait_asynccnt)
    __builtin_amdgcn_s_wait_asynccnt(0);
#else
    asm volatile("s_wait_asynccnt 0" ::: "memory");
#endif
#endif
    __syncthreads();

    // ---- Phase 2: stream all batch*head rows for our 8 positions -------
    const int pl = t >> 5;                       // position 0..7
    const int q  = t & 31;                       // float4 index in row (32*4=128)
    const int s  = blockIdx.x * POS_PER_BLOCK + pl;

    const float c0  = s_cos[pl][2 * q + 0];
    const float c1  = s_cos[pl][2 * q + 1];
    const float sn0 = s_sin[pl][2 * q + 0];
    const float sn1 = s_sin[pl][2 * q + 1];

    const size_t rowbase = (size_t)s * DK + 4 * (size_t)q;
    const size_t bh_stride = (size_t)SEQ * DK;   // 524288 elements

#pragma unroll 4
    for (int bh = 0; bh < BH_TOTAL; ++bh) {
        const size_t off = (size_t)bh * bh_stride + rowbase;
        if (bh + 4 < BH_TOTAL)
            __builtin_prefetch(x + off + 4 * bh_stride, 0, 0);  // global_prefetch

        // x/out are single-use: non-temporal to avoid polluting L2 so the
        // gathered R diagonal lines stay resident for other blocks.
        v4f v = __builtin_nontemporal_load((const v4f*)(x + off));
        v4f o;
        o.x = c0  * v.x - sn0 * v.y;
        o.y = sn0 * v.x + c0  * v.y;
        o.z = c1  * v.z - sn1 * v.w;
        o.w = sn1 * v.z + c1  * v.w;
        __builtin_nontemporal_store(o, (v4f*)(out + off));
    }
}

extern "C" void kernel_launch(void* const* d_in, const int* in_sizes, int n_in,
                              void* d_out, int out_size, void* d_ws, size_t ws_size,
                              hipStream_t stream) {
    const float* x   = (const float*)d_in[0];   // [4,16,4096,128] f32
    const int*   pos = (const int*)d_in[1];     // [4096] i32
    const float* R   = (const float*)d_in[2];   // [4096,128,128] f32
    float* out = (float*)d_out;                 // [4,16,4096,128] f32

    dim3 grid(SEQ / POS_PER_BLOCK);             // 512 blocks
    dim3 block(THREADS);                        // 8 wave32 waves
    rope_apply_kernel<<<grid, block, 0, stream>>>(x, pos, R, out);
}